// Shrink_83442624626826
// MI455X (gfx1250) — compile-verified
//
#include <hip/hip_runtime.h>

#define TT 1024   // T
#define BB 32     // B
#define DD 512    // D
#define CC 1024   // C

static_assert(BB * DD == 16384, "row offset packing assumes B*D == 2^14");

typedef float v2f __attribute__((ext_vector_type(2)));
typedef float v8f __attribute__((ext_vector_type(8)));

// ---------------------------------------------------------------------------
// Kernel 1: per-(t,b) argmax over C=1024 logits. One wave32 per row.
// logit layout: [T, B, C] row-major -> row = t*B + b.
// pred stored [B, T] as int. First-occurrence argmax semantics preserved.
// ---------------------------------------------------------------------------
__global__ __launch_bounds__(256) void k_argmax(const float* __restrict__ logit,
                                                const unsigned char* __restrict__ padding,
                                                int* __restrict__ pred) {
    const int wid  = threadIdx.x >> 5;
    const int lane = threadIdx.x & 31;
    const int row  = blockIdx.x * 8 + wid;          // row = t*B + b
    const int t = row / BB;
    const int b = row % BB;
    const float4* base = (const float4*)(logit + (size_t)row * CC);

    float bv = -__builtin_inff();
    int   bi = 0;
    // lane l covers contiguous float4 chunks; indices strictly increasing per
    // lane, so strict '>' keeps the first occurrence. 512B coalesced per step.
    #pragma unroll
    for (int k = 0; k < CC / 128; ++k) {
        const float4 v = base[lane + k * 32];
        const int i0 = (lane + k * 32) * 4;
        if (v.x > bv) { bv = v.x; bi = i0;     }
        if (v.y > bv) { bv = v.y; bi = i0 + 1; }
        if (v.z > bv) { bv = v.z; bi = i0 + 2; }
        if (v.w > bv) { bv = v.w; bi = i0 + 3; }
    }
    // wave32 reduce: larger value wins; ties -> smaller index
    #pragma unroll
    for (int off = 16; off >= 1; off >>= 1) {
        const float ov = __shfl_xor(bv, off, 32);
        const int   oi = __shfl_xor(bi, off, 32);
        if (ov > bv || (ov == bv && oi < bi)) { bv = ov; bi = oi; }
    }
    if (lane == 0) {
        const int p = padding[b * TT + t] ? 0 : bi;   // padded -> BLANK(0)
        pred[b * TT + t] = p;
    }
}

// ---------------------------------------------------------------------------
// Kernel 2: per-batch compaction + run-length grouping. One block (1024 thr)
// per batch row. Emits packed per-frame metadata {seg|src<<16, bits(w)} for
// kernel 3 and writes out_padding / gloss_fake / new_len directly.
// ---------------------------------------------------------------------------
__global__ __launch_bounds__(1024) void k_scan(const int* __restrict__ pred,
                                               int2*  __restrict__ pk,
                                               int*   __restrict__ segstart,
                                               int*   __restrict__ meta,
                                               float* __restrict__ out_pad,
                                               float* __restrict__ gloss_out,
                                               float* __restrict__ newlen_out) {
    __shared__ int s_scan[TT];
    __shared__ int s_cp[TT];
    __shared__ int s_src[TT];
    __shared__ int s_start[TT + 1];

    const int b = blockIdx.x;
    const int t = threadIdx.x;

    const int p = pred[b * TT + t];
    const int m = (p != 0) ? 1 : 0;

    // --- inclusive scan #1: compaction destinations ---
    s_scan[t] = m; __syncthreads();
    for (int off = 1; off < TT; off <<= 1) {
        const int v = (t >= off) ? s_scan[t - off] : 0;
        __syncthreads();
        s_scan[t] += v;
        __syncthreads();
    }
    const int nb   = s_scan[TT - 1];   // number of non-blank frames
    const int dest = s_scan[t] - 1;

    s_cp[t] = -1; __syncthreads();
    if (m) { s_cp[dest] = p; s_src[dest] = t; }
    __syncthreads();

    const int cp    = s_cp[t];
    const int valid = (cp >= 0);
    const int prevv = (t > 0) ? s_cp[t - 1] : -2;
    const int st    = (valid && cp != prevv) ? 1 : 0;

    // --- inclusive scan #2: segment ids ---
    s_scan[t] = st; __syncthreads();
    for (int off = 1; off < TT; off <<= 1) {
        const int v = (t >= off) ? s_scan[t - off] : 0;
        __syncthreads();
        s_scan[t] += v;
        __syncthreads();
    }
    const int nl  = s_scan[TT - 1];    // new_len: number of runs
    const int seg = s_scan[t] - 1;

    if (st) s_start[seg] = t;          // segment -> first compacted frame
    if (t == 0) s_start[nl] = nb;      // sentinel (disjoint index)
    __syncthreads();

    float w = 0.0f;
    if (valid) w = 1.0f / (float)(s_start[seg + 1] - s_start[seg]);

    // --- packed per-frame metadata for kernel 3 ---
    // low 16 bits: segment id (0xFFFF sentinel never matches a tile row);
    // high 16 bits: source frame t (row elem offset = (x & 0xFFFF0000) >> 2)
    int2 pkv;
    pkv.x = valid ? (seg | (s_src[t] << 16)) : 0xFFFF;
    pkv.y = __float_as_int(w);
    pk[b * TT + t] = pkv;

    int ssv = nb;
    if (t <= nl) ssv = s_start[t];
    segstart[b * (TT + 1) + t] = ssv;
    if (t == 0) segstart[b * (TT + 1) + TT] = nb;

    // --- direct outputs ---
    out_pad[b * TT + t] = (t < nl) ? 0.0f : 1.0f;
    int g = -1;
    if (t < nl) g = s_cp[s_start[t]];
    gloss_out[b * TT + t] = (float)g;
    if (t == 0) {
        newlen_out[b] = (float)nl;
        meta[b * 2 + 0] = nb;
        meta[b * 2 + 1] = nl;
    }
}

// ---------------------------------------------------------------------------
// Kernel 3: segmented-mean einsum via V_WMMA_F32_16X16X4_F32.
// Block = (segment tile of 16, batch). 8 waves; each wave sweeps the tile's
// compacted-frame range ONCE with 4 accumulators (4 d-tiles): the A tile is
// synthesized from a SINGLE ds_load_b128 per K-step (the lane's two frames
// are an even pair) and amortized over 4 WMMAs. The weight words get an
// opaque-use anchor so the compiler cannot predicate their load -> the loop
// is branch-free and EXEC is all-ones at every WMMA.
//
// ISA layouts (wave32):
//   A 16x4 : lane m = l&15, VGPR j holds K = j + 2*(l>>4)
//   B 4x16 : lane n = l&15, VGPR j holds K = j + 2*(l>>4)
//   C 16x16: VGPR r -> row r (lanes 0-15) / row r+8 (lanes 16-31), col = l&15
// ---------------------------------------------------------------------------
__global__ __launch_bounds__(256) void k_avg_wmma(const float* __restrict__ rep,
                                                  const int2*  __restrict__ pk,
                                                  const int*   __restrict__ segstart,
                                                  const int*   __restrict__ meta,
                                                  float* __restrict__ out) {
    __shared__ int2 lpk[TT + 4];

    const int b  = blockIdx.y;
    const int n0 = blockIdx.x * 16;
    const int nl = meta[b * 2 + 1];

    const int lo = min(n0, nl);
    const int hi = min(n0 + 16, nl);
    const int kstart = segstart[b * (TT + 1) + lo];
    const int kend   = segstart[b * (TT + 1) + hi];
    const int L      = kend - kstart;

    // stage packed frame metadata + zero-weight sentinel pad
    for (int k = kstart + (int)threadIdx.x; k < kend; k += blockDim.x)
        lpk[k - kstart] = pk[b * TT + k];
    if (threadIdx.x < 4) lpk[L + threadIdx.x] = make_int2(0xFFFF, 0);
    __syncthreads();

    const int lane  = threadIdx.x & 31;
    const int wid   = threadIdx.x >> 5;
    const int l15   = lane & 15;          // A row m / B col n / store col
    const int khalf = lane >> 4;
    const int segwant = n0 + l15;

    int dcol[4];
    #pragma unroll
    for (int dti = 0; dti < 4; ++dti) dcol[dti] = (wid + dti * 8) * 16 + l15;

    const float* __restrict__ repb = rep + b * DD;    // uniform base
    const int4*  __restrict__ lpk4 = (const int4*)lpk;

    v8f acc[4];
    #pragma unroll
    for (int dti = 0; dti < 4; ++dti) acc[dti] = (v8f){};

    for (int k0 = 0; k0 < L; k0 += 4) {               // uniform per block
        // one b128 LDS load: metadata for this lane's frame pair (even index)
        const int4 mm = lpk4[(k0 >> 1) + khalf];

        int w0 = mm.y, w1 = mm.w;
        asm volatile("" : "+v"(w0), "+v"(w1));        // forbid predicated load

        const int seg0 = mm.x & 0xFFFF;
        const int seg1 = mm.z & 0xFFFF;
        const int off0 = (mm.x & 0xFFFF0000) >> 2;    // src0 * B*D (2^14)
        const int off1 = (mm.z & 0xFFFF0000) >> 2;    // src1 * B*D

        v2f a;
        a[0] = (seg0 == segwant) ? __int_as_float(w0) : 0.0f;
        a[1] = (seg1 == segwant) ? __int_as_float(w1) : 0.0f;

        #pragma unroll
        for (int dti = 0; dti < 4; ++dti) {
            v2f bv;
            bv[0] = repb[off0 + dcol[dti]];
            bv[1] = repb[off1 + dcol[dti]];
            acc[dti] = __builtin_amdgcn_wmma_f32_16x16x4_f32(
                false, a, false, bv, (short)0, acc[dti], false, false);
        }
    }

    #pragma unroll
    for (int dti = 0; dti < 4; ++dti) {
        #pragma unroll
        for (int r = 0; r < 8; ++r) {
            const int nrow = n0 + r + 8 * khalf;
            out[((size_t)nrow * BB + b) * DD + dcol[dti]] = acc[dti][r];
        }
    }
}

// ---------------------------------------------------------------------------
extern "C" void kernel_launch(void* const* d_in, const int* in_sizes, int n_in,
                              void* d_out, int out_size, void* d_ws, size_t ws_size,
                              hipStream_t stream) {
    const float*         rep     = (const float*)d_in[0];         // [T,B,D]
    const float*         logit   = (const float*)d_in[1];         // [T,B,C]
    const unsigned char* padding = (const unsigned char*)d_in[2]; // [B,T] bool

    // workspace layout
    int*  pred     = (int*)d_ws;                        // B*T ints
    int*  segstart = pred + BB * TT;                    // B*(T+1) ints
    int*  meta     = segstart + BB * (TT + 1);          // B*2 ints
    size_t pk_off  = ((size_t)(meta + BB * 2 - (int*)d_ws) * 4 + 15) & ~(size_t)15;
    int2* pk       = (int2*)((char*)d_ws + pk_off);     // B*T int2 (16B-aligned)

    // output layout (all float32): out [T,B,D] | out_padding [B,T] |
    // gloss_fake [B,T] | new_len [B]
    float* out        = (float*)d_out;
    float* out_pad    = out + (size_t)TT * BB * DD;
    float* gloss_out  = out_pad + BB * TT;
    float* newlen_out = gloss_out + BB * TT;

    // K1: argmax, 8 wave32 rows per 256-thread block
    k_argmax<<<dim3((TT * BB) / 8), dim3(256), 0, stream>>>(logit, padding, pred);

    // K2: per-batch scans
    k_scan<<<dim3(BB), dim3(TT), 0, stream>>>(pred, pk, segstart, meta,
                                              out_pad, gloss_out, newlen_out);

    // K3: WMMA segmented mean, grid = (segment tiles, batches)
    k_avg_wmma<<<dim3(TT / 16, BB), dim3(256), 0, stream>>>(rep, pk, segstart,
                                                            meta, out);
}